// MGKBertSelfAttention_8916352106651
// MI455X (gfx1250) — compile-verified
//
#include <hip/hip_runtime.h>
#include <math.h>

typedef float v2f __attribute__((ext_vector_type(2)));
typedef float v8f __attribute__((ext_vector_type(8)));

// D = A(16x4) * B(4x16) + C, all f32.
// (neg_a, A, neg_b, B, c_mod, C, reuse_a, reuse_b)
#define WMMA_F32(a, b, c) \
    __builtin_amdgcn_wmma_f32_16x16x4_f32(false, (a), false, (b), (short)0, (c), false, false)

static constexpr int Bb = 4, Ss = 1024, HIDc = 768, Hh = 12, Dd = 32, AHc = 384;
static constexpr float C1C = 0.08838834764831845f;   // SCALING/2 = 0.5/sqrt(32)
static constexpr float C2C = 0.13258252147247767f;   // 1.5*SCALING/2

// CDNA5 async global->LDS copy: 16 bytes per lane, no VGPR round trip,
// tracked with ASYNCcnt (CDNA5 ISA).
__device__ __forceinline__ void async_copy_b128(unsigned lds_off, const void* gaddr)
{
    asm volatile("global_load_async_to_lds_b128 %0, %1, off"
                 :: "v"(lds_off), "v"(gaddr)
                 : "memory");
}
__device__ __forceinline__ void wait_asynccnt0()
{
    asm volatile("s_wait_asynccnt 0x0" ::: "memory");
}
// Low 32 bits of a generic pointer into shared memory == LDS byte offset
// within the workgroup allocation (ISA §10.2 aperture mapping).
__device__ __forceinline__ unsigned lds_offset_of(const void* p)
{
    return (unsigned)(unsigned long long)p;
}

// ---------------------------------------------------------------------------
// Kernel 1: QKV projections.  out = X @ W^T + b, written as [B,H,S,D].
// One wave per 16x16 output tile; K=768 -> 192 v_wmma_f32_16x16x4_f32.
// ---------------------------------------------------------------------------
__global__ __launch_bounds__(32)
void proj_kernel(const float* __restrict__ X,
                 const float* __restrict__ Wq,  const float* __restrict__ bq,
                 const float* __restrict__ Wk1, const float* __restrict__ bk1,
                 const float* __restrict__ Wk2, const float* __restrict__ bk2,
                 const float* __restrict__ Wv,  const float* __restrict__ bv,
                 float* __restrict__ Qo, float* __restrict__ K1o,
                 float* __restrict__ K2o, float* __restrict__ Vo)
{
    const int lane = threadIdx.x & 31;
    const int lm = lane & 15, lh = lane >> 4;
    const int r0 = blockIdx.x * 16;      // row in [B*S]
    const int c0 = blockIdx.y * 16;      // col in [AH]
    const int sel = blockIdx.z;

    const float* W; const float* bias; float* out;
    switch (sel) {
        case 0:  W = Wq;  bias = bq;  out = Qo;  break;
        case 1:  W = Wk1; bias = bk1; out = K1o; break;
        case 2:  W = Wk2; bias = bk2; out = K2o; break;
        default: W = Wv;  bias = bv;  out = Vo;  break;
    }

    // A lane: row = lm, k = base + 2*lh + {0,1}.  B lane: col = lm, same k map.
    const float* ap = X + (size_t)(r0 + lm) * HIDc + 2 * lh;
    const float* bp = W + (size_t)(c0 + lm) * HIDc + 2 * lh;

    v8f c = {};
    for (int k = 0; k < HIDc; k += 4) {
        v2f a, b;
        a.x = ap[k]; a.y = ap[k + 1];
        b.x = bp[k]; b.y = bp[k + 1];
        c = WMMA_F32(a, b, c);
    }

    const int bidx = r0 >> 10;           // batch
    const int srow = r0 & 1023;          // seq pos within batch (16-aligned)
    const int h  = c0 >> 5;              // head
    const int d0 = c0 & 31;              // 0 or 16 within head
    const float biasv = bias[c0 + lm];
    float* obase = out + (((size_t)bidx * Hh + h) * Ss + srow) * Dd + d0 + lm;
    #pragma unroll
    for (int v = 0; v < 8; ++v) {
        const int m = v + 8 * lh;        // D-matrix row for this vgpr/lane-half
        obase[(size_t)m * Dd] = c[v] + biasv;
    }
}

// ---------------------------------------------------------------------------
// Kernel 2: adj[bh][d] = mean over S of Q  (gpytorch mean-adjustment; the
// same adj is applied to q, k1 and k2 since _sq_dist derives it from x1=q).
// ---------------------------------------------------------------------------
__global__ __launch_bounds__(256)
void adj_kernel(const float* __restrict__ Q, float* __restrict__ adj)
{
    __shared__ float part[256];
    const int bh = blockIdx.x;
    const int tid = threadIdx.x;
    const int d = tid & 31, g = tid >> 5;
    float s = 0.f;
    for (int srow = g; srow < Ss; srow += 8)
        s += Q[((size_t)bh * Ss + srow) * Dd + d];
    part[tid] = s;
    __syncthreads();
    if (tid < 32) {
        float t = 0.f;
        for (int gg = 0; gg < 8; ++gg) t += part[gg * 32 + tid];
        adj[bh * Dd + tid] = t * (1.0f / 1024.0f);
    }
}

// ---------------------------------------------------------------------------
// Kernel 3: fused MGK attention.  One block (4 waves) per (b,h, 16 q-rows).
// Both full score planes (2 x 16x1024 f32 = 128KB) live in LDS so probs is
// written to HBM exactly once (201MB is the roofline-dominant traffic).
// K tiles are staged RAW via global_load_async_to_lds_b128 (ASYNCcnt path);
// the mean-adjustment is folded algebraically:
//   (q-adj).(k-adj) = (q-adj).k - (q-adj).adj  = cross_raw - t[m]
// ---------------------------------------------------------------------------
__global__ __launch_bounds__(128)
void attn_kernel(const float* __restrict__ Q,  const float* __restrict__ K1,
                 const float* __restrict__ K2, const float* __restrict__ V,
                 const float* __restrict__ adj, const float* __restrict__ mask,
                 const float* __restrict__ pi,
                 float* __restrict__ ctx_out, float* __restrict__ probs_out)
{
    extern __shared__ float sm[];
    float* qs   = sm;            // 512   adjusted q tile [16][32]
    float* nqv  = sm + 512;      // 16    ||q-adj||^2 per row
    float* tqv  = sm + 528;      // 16    (q-adj).adj per row
    float* nk1  = sm + 544;      // 64    per-wave k1 row norms (adjusted)
    float* nk2  = sm + 608;      // 64
    float* m1w  = sm + 672;      // 64    per-wave row maxes
    float* m2w  = sm + 736;      // 64
    float* m1s  = sm + 800;      // 16
    float* m2s  = sm + 816;      // 16
    float* rsum = sm + 832;      // 128
    float* invs = sm + 960;      // 16
    float* ctxp = sm + 1024;     // 2048  per-wave ctx partials [4][16][32]
    float* s1   = sm + 3072;     // 16384 score plane 1 [16][1024]
    float* s2   = sm + 19456;    // 16384 score plane 2
    float* kst1 = sm + 35840;    // 2048  staged RAW k1 [4][16][32]
    float* kst2 = sm + 37888;    // 2048  total: 39936 floats = 156KB

    const int tid = threadIdx.x;
    const int w = tid >> 5, lane = tid & 31;
    const int lm = lane & 15, lh = lane >> 4;
    const int bh = blockIdx.y, b = bh / Hh, h = bh % Hh;
    const int s0 = blockIdx.x * 16;

    const float* adjv = adj + bh * Dd;
    const float* qb   = Q  + ((size_t)bh * Ss + s0) * Dd;
    const float* k1b  = K1 + (size_t)bh * Ss * Dd;
    const float* k2b  = K2 + (size_t)bh * Ss * Dd;
    const float* vb   = V  + (size_t)bh * Ss * Dd;

    const unsigned kst1_base = lds_offset_of(kst1);
    const unsigned kst2_base = lds_offset_of(kst2);

    for (int i = tid; i < 512; i += 128)
        qs[i] = qb[i] - adjv[i & 31];
    __syncthreads();
    if (tid < 16) {
        float n = 0.f, t = 0.f;
        for (int d = 0; d < Dd; ++d) {
            const float x = qs[tid * Dd + d];
            n += x * x;
            t += x * adjv[d];
        }
        nqv[tid] = n;
        tqv[tid] = t;
    }
    const float p1 = fminf(fmaxf(pi[h], 1e-6f), 2.0f);
    const float p2 = fminf(fmaxf(pi[Hh + h], 1e-6f), 2.0f);
    __syncthreads();

    float mx1[8], mx2[8];
    #pragma unroll
    for (int v = 0; v < 8; ++v) { mx1[v] = -3.0e38f; mx2[v] = -3.0e38f; }

    // ---- Phase 1: both score planes + running row maxes -------------------
    for (int it = 0; it < 16; ++it) {
        // Async-DMA 4 raw k1 tiles + 4 raw k2 tiles into LDS (16B per issue).
        // Chunk i covers floats [4i, 4i+4): tile wv = i>>7, row = (i&127)>>3,
        // d = (i&7)*4 -> contiguous within a k-row, so each chunk is 16B of
        // the row-major [B,H,S,D] buffer.
        const int jbase = it * 64;          // first k row staged this iter
        for (int i = tid; i < 512; i += 128) {
            const size_t foff = (size_t)(jbase * Dd) + (size_t)i * 4;
            async_copy_b128(kst1_base + (unsigned)i * 16u, k1b + foff);
            async_copy_b128(kst2_base + (unsigned)i * 16u, k2b + foff);
        }
        wait_asynccnt0();
        __syncthreads();
        if (tid < 64) {                                   // k1 row norms (adj)
            const int wv = tid >> 4, r = tid & 15;
            const float* p = kst1 + wv * 512 + r * Dd;
            float s = 0.f;
            for (int d = 0; d < Dd; ++d) {
                const float x = p[d] - adjv[d];
                s += x * x;
            }
            nk1[wv * 16 + r] = s;
        } else {                                          // k2 row norms (adj)
            const int t2 = tid - 64, wv = t2 >> 4, r = t2 & 15;
            const float* p = kst2 + wv * 512 + r * Dd;
            float s = 0.f;
            for (int d = 0; d < Dd; ++d) {
                const float x = p[d] - adjv[d];
                s += x * x;
            }
            nk2[wv * 16 + r] = s;
        }
        __syncthreads();

        const int j0 = (it * 4 + w) * 16;
        const float* myk1 = kst1 + w * 512;
        const float* myk2 = kst2 + w * 512;
        v8f c1 = {}, c2 = {};
        #pragma unroll
        for (int kk = 0; kk < Dd; kk += 4) {
            const int ko = kk + 2 * lh;
            v2f a, bb1, bb2;
            a.x   = qs[lm * Dd + ko];    a.y   = qs[lm * Dd + ko + 1];
            bb1.x = myk1[lm * Dd + ko];  bb1.y = myk1[lm * Dd + ko + 1];
            bb2.x = myk2[lm * Dd + ko];  bb2.y = myk2[lm * Dd + ko + 1];
            c1 = WMMA_F32(a, bb1, c1);
            c2 = WMMA_F32(a, bb2, c2);
        }
        const float maskv = mask[(size_t)b * Ss + j0 + lm];
        const float nk1v = nk1[w * 16 + lm];
        const float nk2v = nk2[w * 16 + lm];
        #pragma unroll
        for (int v = 0; v < 8; ++v) {
            const int m = v + 8 * lh;
            const float nq = nqv[m];
            const float cadj1 = c1[v] - tqv[m];   // (q-adj).(k1-adj)
            const float cadj2 = c2[v] - tqv[m];   // (q-adj).(k2-adj)
            const float d1 = -C1C * fmaxf(nq + nk1v - 2.0f * cadj1, 0.f) + maskv;
            const float d2 = -C2C * fmaxf(nq + nk2v - 2.0f * cadj2, 0.f) + maskv;
            s1[(size_t)m * Ss + j0 + lm] = d1;
            s2[(size_t)m * Ss + j0 + lm] = d2;
            mx1[v] = fmaxf(mx1[v], d1);
            mx2[v] = fmaxf(mx2[v], d2);
        }
        __syncthreads();
    }

    // row-max reduce: lanes 0-15 / 16-31 each share a row set
    #pragma unroll
    for (int off = 8; off >= 1; off >>= 1) {
        #pragma unroll
        for (int v = 0; v < 8; ++v) {
            mx1[v] = fmaxf(mx1[v], __shfl_xor(mx1[v], off, 32));
            mx2[v] = fmaxf(mx2[v], __shfl_xor(mx2[v], off, 32));
        }
    }
    if (lm == 0) {
        #pragma unroll
        for (int v = 0; v < 8; ++v) {
            m1w[w * 16 + v + 8 * lh] = mx1[v];
            m2w[w * 16 + v + 8 * lh] = mx2[v];
        }
    }
    __syncthreads();
    if (tid < 16) {
        float mm = m1w[tid];
        for (int ww = 1; ww < 4; ++ww) mm = fmaxf(mm, m1w[ww * 16 + tid]);
        m1s[tid] = mm;
    } else if (tid < 32) {
        const int r = tid - 16;
        float mm = m2w[r];
        for (int ww = 1; ww < 4; ++ww) mm = fmaxf(mm, m2w[ww * 16 + r]);
        m2s[r] = mm;
    }
    __syncthreads();

    // ---- Phase 2: exp-mix, row sums ---------------------------------------
    {
        const int r = tid >> 3, cidx = tid & 7;
        const float mm1 = m1s[r], mm2 = m2s[r];
        float rs = 0.f;
        for (int j = cidx; j < Ss; j += 8) {
            const float e = __expf(s1[r * Ss + j] - mm1) * p1
                          + __expf(s2[r * Ss + j] - mm2) * p2;
            s1[r * Ss + j] = e;                 // unnormalized mixed scores
            rs += e;
        }
        rsum[r * 8 + cidx] = rs;
    }
    __syncthreads();
    if (tid < 16) {
        float s = 0.f;
        for (int g = 0; g < 8; ++g) s += rsum[tid * 8 + g];
        invs[tid] = 1.0f / (s + 1e-6f);
    }
    __syncthreads();

    // single normalized probs write; non-temporal (201MB >> keeps L2 for QKV)
    {
        float* pout = probs_out + ((size_t)bh * Ss + s0) * Ss;
        for (int i = tid; i < 16 * Ss; i += 128)
            __builtin_nontemporal_store(s1[i] * invs[i >> 10], pout + i);
    }

    // ---- Phase 3: ctx = (scores @ V) * inv, K split across 4 waves --------
    v8f cc0 = {}, cc1 = {};
    const int k0 = w * 256;
    for (int kk = k0; kk < k0 + 256; kk += 4) {
        const int kr = kk + 2 * lh;
        v2f a, b0, b1;
        a.x  = s1[lm * Ss + kr];               a.y  = s1[lm * Ss + kr + 1];
        b0.x = vb[(size_t)kr * Dd + lm];       b0.y = vb[(size_t)(kr + 1) * Dd + lm];
        b1.x = vb[(size_t)kr * Dd + 16 + lm];  b1.y = vb[(size_t)(kr + 1) * Dd + 16 + lm];
        cc0 = WMMA_F32(a, b0, cc0);
        cc1 = WMMA_F32(a, b1, cc1);
    }
    #pragma unroll
    for (int v = 0; v < 8; ++v) {
        const int m = v + 8 * lh;
        ctxp[w * 512 + m * Dd + lm]      = cc0[v];
        ctxp[w * 512 + m * Dd + 16 + lm] = cc1[v];
    }
    __syncthreads();
    for (int i = tid; i < 512; i += 128) {
        const int m = i >> 5, d = i & 31;
        const float acc = ctxp[i] + ctxp[512 + i] + ctxp[1024 + i] + ctxp[1536 + i];
        __builtin_nontemporal_store(acc * invs[m],
            ctx_out + ((size_t)b * Ss + s0 + m) * AHc + h * Dd + d);
    }
}

// ---------------------------------------------------------------------------
extern "C" void kernel_launch(void* const* d_in, const int* in_sizes, int n_in,
                              void* d_out, int out_size, void* d_ws, size_t ws_size,
                              hipStream_t stream)
{
    const float* hidden = (const float*)d_in[0];
    const float* mask   = (const float*)d_in[1];
    const float* Wq  = (const float*)d_in[2];
    const float* bq  = (const float*)d_in[3];
    const float* Wk1 = (const float*)d_in[4];
    const float* bk1 = (const float*)d_in[5];
    const float* Wk2 = (const float*)d_in[6];
    const float* bk2 = (const float*)d_in[7];
    const float* Wv  = (const float*)d_in[8];
    const float* bv  = (const float*)d_in[9];
    const float* pi  = (const float*)d_in[10];

    float* ws = (float*)d_ws;
    const size_t per = (size_t)Bb * Hh * Ss * Dd;   // 1,572,864 floats
    float* Qd   = ws;
    float* K1d  = ws + per;
    float* K2d  = ws + 2 * per;
    float* Vd   = ws + 3 * per;
    float* adjd = ws + 4 * per;                     // + 1536 floats

    float* ctx_out   = (float*)d_out;                          // [B,S,AH]
    float* probs_out = ctx_out + (size_t)Bb * Ss * AHc;        // [B,H,S,S]

    constexpr size_t SMEM = 39936 * sizeof(float);  // 156 KB dynamic LDS
    (void)hipFuncSetAttribute(reinterpret_cast<const void*>(attn_kernel),
                              hipFuncAttributeMaxDynamicSharedMemorySize,
                              (int)SMEM);

    proj_kernel<<<dim3(256, 24, 4), 32, 0, stream>>>(
        hidden, Wq, bq, Wk1, bk1, Wk2, bk2, Wv, bv, Qd, K1d, K2d, Vd);
    adj_kernel<<<dim3(48), 256, 0, stream>>>(Qd, adjd);
    attn_kernel<<<dim3(64, 48), 128, SMEM, stream>>>(
        Qd, K1d, K2d, Vd, adjd, mask, pi, ctx_out, probs_out);
}